// LDMK_Conv_28870770164151
// MI455X (gfx1250) — compile-verified
//
#include <hip/hip_runtime.h>
#include <hip/hip_bf16.h>
#include <math.h>

// ---------------------------------------------------------------------------
// LDMK_Conv for MI455X (gfx1250, wave32).
// B=8, Cin=M=O=256, H=W=128, k=128.  All fp32 -> V_WMMA_F32_16X16X4_F32.
// ---------------------------------------------------------------------------

typedef __attribute__((ext_vector_type(2))) float v2f;
typedef __attribute__((ext_vector_type(8))) float v8f;

#define BATCH 8
#define CIN   256
#define MCH   256
#define KSEL  128
#define HH    128
#define WW    128
#define HWPX  (HH * WW)          // 16384
#define FCH   (3 * KSEL)         // 384 fused channels
#define NTAPS 83                 // 9 + 25 + 49

__device__ __forceinline__ float sigmoidf_(float x) { return 1.f / (1.f + __expf(-x)); }

// ---------------------------------------------------------------------------
// K0: zero the gap accumulators (4096 floats: gapsum | gap2sum)
// ---------------------------------------------------------------------------
__global__ void k_init(float* __restrict__ p) {
    p[blockIdx.x * 256 + threadIdx.x] = 0.f;
}

// ---------------------------------------------------------------------------
// K1: h = pw1_w @ x + b   (per-batch 256x256x16384 GEMM via WMMA f32 16x16x4)
//     fused: atomic accumulation of sum(h) and sum(relu(h)) per (b, m).
// Block = 256 threads = 8 waves. Block tile: 256 M x 32 pixels.
//   wave&3  -> 64-row M chunk (4 accumulators, B reused across them)
//   wave>>2 -> 16-pixel column chunk
// Grid = (HW/32, B)
// ---------------------------------------------------------------------------
__global__ __launch_bounds__(256) void k_pw1(
    const float* __restrict__ x, const float* __restrict__ w,
    const float* __restrict__ bias, float* __restrict__ h,
    float* __restrict__ gapsum, float* __restrict__ gap2sum) {
    const int b    = blockIdx.y;
    const int lane = threadIdx.x & 31;
    const int wv   = threadIdx.x >> 5;
    const int l15  = lane & 15;
    const int hi   = lane >> 4;                    // 0 or 1 (K-half / M-half)
    const int mB   = (wv & 3) * 64;
    const int pix  = blockIdx.x * 32 + (wv >> 2) * 16 + l15;   // N = l15

    const float* __restrict__ xb = x + (size_t)b * CIN * HWPX;

    v8f acc[4] = {};
    for (int kk = 0; kk < CIN; kk += 4) {
        const int kA = kk + 2 * hi;                // this lane's K base
        v2f bm;                                    // B 4x16: lane N=l15, K=kA,kA+1
        bm.x = xb[(size_t)kA * HWPX + pix];
        bm.y = xb[(size_t)(kA + 1) * HWPX + pix];
#pragma unroll
        for (int mi = 0; mi < 4; ++mi) {
            const int m = mB + mi * 16 + l15;      // A 16x4: lane M=l15, K=kA,kA+1
            v2f am;
            am.x = w[m * CIN + kA];
            am.y = w[m * CIN + kA + 1];
            acc[mi] = __builtin_amdgcn_wmma_f32_16x16x4_f32(
                false, am, false, bm, (short)0, acc[mi], false, false);
        }
    }

    // D layout: vgpr r, lane -> N = l15, M = r + 8*hi
#pragma unroll
    for (int mi = 0; mi < 4; ++mi) {
#pragma unroll
        for (int r = 0; r < 8; ++r) {
            const int m = mB + mi * 16 + 8 * hi + r;
            const float v = acc[mi][r] + bias[m];
            h[((size_t)b * MCH + m) * HWPX + pix] = v;
            // reduce the 16 pixels (N lanes) of this row, then one atomic
            float s1 = v, s2 = fmaxf(v, 0.f);
#pragma unroll
            for (int o = 1; o < 16; o <<= 1) {
                s1 += __shfl_xor(s1, o, 32);
                s2 += __shfl_xor(s2, o, 32);
            }
            if (l15 == 0) {
                atomicAdd(&gapsum[b * MCH + m], s1);
                atomicAdd(&gap2sum[b * MCH + m], s2);
            }
        }
    }
}

// ---------------------------------------------------------------------------
// K2: control path (single block): gaps -> fc1(relu) -> fc2(sigmoid) ->
//     top-k membership gate for channels < 128, offsets, modulated dw taps.
// ---------------------------------------------------------------------------
__global__ __launch_bounds__(256) void k_ctrl(
    const float* __restrict__ gapsum, const float* __restrict__ gap2sum,
    const float* __restrict__ fc1_w, const float* __restrict__ fc1_b,
    const float* __restrict__ fc2_w, const float* __restrict__ fc2_b,
    const float* __restrict__ off_w, const float* __restrict__ off_b,
    const float* __restrict__ dw3, const float* __restrict__ dw5,
    const float* __restrict__ dw7,
    float* __restrict__ gate, float* __restrict__ wmod) {
    __shared__ float gap[BATCH * MCH];
    __shared__ float gap2[BATCH * MCH];
    __shared__ float hid[BATCH * 2 * MCH];
    __shared__ float scores[BATCH * MCH];
    __shared__ float offs[BATCH * 6];
    const int t = threadIdx.x;

    for (int i = t; i < BATCH * MCH; i += 256) {
        gap[i]  = gapsum[i]  * (1.f / (float)HWPX);
        gap2[i] = gap2sum[i] * (1.f / (float)HWPX);
    }
    __syncthreads();

    for (int i = t; i < BATCH * 2 * MCH; i += 256) {        // hid = relu(gap @ fc1^T + b)
        const int b = i >> 9, o = i & 511;
        float s = fc1_b[o];
        for (int c = 0; c < MCH; ++c) s += gap[b * MCH + c] * fc1_w[o * MCH + c];
        hid[i] = fmaxf(s, 0.f);
    }
    __syncthreads();

    for (int i = t; i < BATCH * MCH; i += 256) {            // scores = sigmoid(hid @ fc2^T + b)
        const int b = i >> 8, o = i & 255;
        float s = fc2_b[o];
        for (int c = 0; c < 2 * MCH; ++c) s += hid[b * 2 * MCH + c] * fc2_w[o * 2 * MCH + c];
        scores[i] = sigmoidf_(s);
    }
    __syncthreads();

    // top-k membership (k = 128) via rank count; ties broken by lower index
    for (int i = t; i < BATCH * KSEL; i += 256) {
        const int b = i >> 7, c = i & 127;
        const float s = scores[b * MCH + c];
        int rank = 0;
        for (int j = 0; j < MCH; ++j) {
            const float sj = scores[b * MCH + j];
            rank += (sj > s) || (sj == s && j < c);
        }
        gate[i] = (rank < KSEL) ? 1.f : 0.f;
    }

    if (t < BATCH * 6) {                                    // offsets = tanh(gap2 @ off^T + b)
        const int b = t / 6, o = t % 6;
        float s = off_b[o];
        for (int c = 0; c < MCH; ++c) s += gap2[b * MCH + c] * off_w[o * MCH + c];
        offs[t] = tanhf(s);
    }
    __syncthreads();

    // modulated depthwise taps: wmod[(b*128+c)*83 + tap]; taps 0-8:3x3, 9-33:5x5, 34-82:7x7
    for (int i = t; i < BATCH * KSEL * NTAPS; i += 256) {
        const int tap = i % NTAPS;
        const int bc = i / NTAPS;
        const int b = bc >> 7, c = bc & 127;
        int branch, local;
        const float* src;
        if (tap < 9)       { branch = 0; local = tap;      src = dw3 + c * 9; }
        else if (tap < 34) { branch = 1; local = tap - 9;  src = dw5 + c * 25; }
        else               { branch = 2; local = tap - 34; src = dw7 + c * 49; }
        const float scale = sigmoidf_(offs[b * 6 + branch * 2]);
        const float shift = tanhf(offs[b * 6 + branch * 2 + 1]);
        wmod[i] = src[local] * (1.f + scale) + shift;
    }
}

// ---------------------------------------------------------------------------
// K3: three depthwise convs from a single shared halo tile.
// Block: (b, c) x 32x32 tile; LDS tile 38x38 (~5.8 KB).  Input = relu(h)*gate.
// Grid = (16 tiles, B*128)
// ---------------------------------------------------------------------------
#define TS 32
#define HALO 3
__global__ __launch_bounds__(256) void k_dw(
    const float* __restrict__ h, const float* __restrict__ gate,
    const float* __restrict__ wmod, float* __restrict__ fused) {
    const int bc = blockIdx.y;
    const int b = bc >> 7, c = bc & 127;
    const int tx0 = (blockIdx.x & 3) * TS;
    const int ty0 = (blockIdx.x >> 2) * TS;

    __shared__ float tile[TS + 2 * HALO][TS + 2 * HALO];
    __shared__ float wt[NTAPS];

    const float g = gate[bc];
    const float* __restrict__ hp = h + ((size_t)b * MCH + c) * HWPX;

    for (int i = threadIdx.x; i < (TS + 6) * (TS + 6); i += 256) {
        const int ly = i / (TS + 6), lx = i % (TS + 6);
        const int gy = ty0 + ly - HALO, gx = tx0 + lx - HALO;
        float v = 0.f;
        if (gy >= 0 && gy < HH && gx >= 0 && gx < WW)
            v = fmaxf(hp[gy * WW + gx], 0.f) * g;
        tile[ly][lx] = v;
    }
    if (threadIdx.x < NTAPS) wt[threadIdx.x] = wmod[(size_t)bc * NTAPS + threadIdx.x];
    __syncthreads();

    for (int p = threadIdx.x; p < TS * TS; p += 256) {
        const int py = p / TS, px = p % TS;
        float a3 = 0.f, a5 = 0.f, a7 = 0.f;
#pragma unroll
        for (int dy = 0; dy < 7; ++dy) {
#pragma unroll
            for (int dx = 0; dx < 7; ++dx) {
                const float v = tile[py + dy][px + dx];
                a7 += v * wt[34 + dy * 7 + dx];
                if (dy >= 1 && dy < 6 && dx >= 1 && dx < 6)
                    a5 += v * wt[9 + (dy - 1) * 5 + (dx - 1)];
                if (dy >= 2 && dy < 5 && dx >= 2 && dx < 5)
                    a3 += v * wt[(dy - 2) * 3 + (dx - 2)];
            }
        }
        const size_t pix = (size_t)(ty0 + py) * WW + (tx0 + px);
        fused[((size_t)b * FCH + c) * HWPX + pix]             = a3;
        fused[((size_t)b * FCH + KSEL + c) * HWPX + pix]      = a5;
        fused[((size_t)b * FCH + 2 * KSEL + c) * HWPX + pix]  = a7;
    }
}

// ---------------------------------------------------------------------------
// K4: out = pw_w[:, :384] @ fused + pw_b + x   (WMMA f32, K = 384)
// pw_w row stride is 768 (shape [256, 3*M]); only first 384 cols used.
// ---------------------------------------------------------------------------
__global__ __launch_bounds__(256) void k_pw2(
    const float* __restrict__ fused, const float* __restrict__ w,
    const float* __restrict__ bias, const float* __restrict__ x,
    float* __restrict__ out) {
    const int b    = blockIdx.y;
    const int lane = threadIdx.x & 31;
    const int wv   = threadIdx.x >> 5;
    const int l15  = lane & 15;
    const int hi   = lane >> 4;
    const int mB   = (wv & 3) * 64;
    const int pix  = blockIdx.x * 32 + (wv >> 2) * 16 + l15;

    const float* __restrict__ fb = fused + (size_t)b * FCH * HWPX;

    v8f acc[4] = {};
    for (int kk = 0; kk < FCH; kk += 4) {
        const int kA = kk + 2 * hi;
        v2f bm;
        bm.x = fb[(size_t)kA * HWPX + pix];
        bm.y = fb[(size_t)(kA + 1) * HWPX + pix];
#pragma unroll
        for (int mi = 0; mi < 4; ++mi) {
            const int m = mB + mi * 16 + l15;
            v2f am;
            am.x = w[m * (3 * MCH) + kA];
            am.y = w[m * (3 * MCH) + kA + 1];
            acc[mi] = __builtin_amdgcn_wmma_f32_16x16x4_f32(
                false, am, false, bm, (short)0, acc[mi], false, false);
        }
    }

#pragma unroll
    for (int mi = 0; mi < 4; ++mi) {
#pragma unroll
        for (int r = 0; r < 8; ++r) {
            const int m = mB + mi * 16 + 8 * hi + r;
            const size_t idx = ((size_t)b * MCH + m) * HWPX + pix;
            out[idx] = acc[mi][r] + bias[m] + x[idx];
        }
    }
}

// ---------------------------------------------------------------------------
// Launch
// ---------------------------------------------------------------------------
extern "C" void kernel_launch(void* const* d_in, const int* in_sizes, int n_in,
                              void* d_out, int out_size, void* d_ws, size_t ws_size,
                              hipStream_t stream) {
    const float* x     = (const float*)d_in[0];
    const float* pw1_w = (const float*)d_in[1];
    const float* pw1_b = (const float*)d_in[2];
    const float* fc1_w = (const float*)d_in[3];
    const float* fc1_b = (const float*)d_in[4];
    const float* fc2_w = (const float*)d_in[5];
    const float* fc2_b = (const float*)d_in[6];
    const float* off_w = (const float*)d_in[7];
    const float* off_b = (const float*)d_in[8];
    const float* dw3   = (const float*)d_in[9];
    const float* dw5   = (const float*)d_in[10];
    const float* dw7   = (const float*)d_in[11];
    const float* pw_w  = (const float*)d_in[12];
    const float* pw_b  = (const float*)d_in[13];
    (void)in_sizes; (void)n_in; (void)out_size; (void)ws_size;   // k hardcoded = 128

    float* ws      = (float*)d_ws;
    float* h       = ws;                                     // 8*256*16384
    float* fusedb  = h + (size_t)BATCH * MCH * HWPX;         // 8*384*16384
    float* gapsum  = fusedb + (size_t)BATCH * FCH * HWPX;    // 2048
    float* gap2sum = gapsum + BATCH * MCH;                   // 2048
    float* gate    = gap2sum + BATCH * MCH;                  // 1024
    float* wmod    = gate + BATCH * KSEL;                    // 8*128*83
    float* out     = (float*)d_out;

    k_init<<<dim3(16), dim3(256), 0, stream>>>(gapsum);      // zeros gapsum+gap2sum
    k_pw1<<<dim3(HWPX / 32, BATCH), dim3(256), 0, stream>>>(
        x, pw1_w, pw1_b, h, gapsum, gap2sum);
    k_ctrl<<<dim3(1), dim3(256), 0, stream>>>(
        gapsum, gap2sum, fc1_w, fc1_b, fc2_w, fc2_b,
        off_w, off_b, dw3, dw5, dw7, gate, wmod);
    k_dw<<<dim3(16, BATCH * KSEL), dim3(256), 0, stream>>>(h, gate, wmod, fusedb);
    k_pw2<<<dim3(HWPX / 32, BATCH), dim3(256), 0, stream>>>(fusedb, pw_w, pw_b, x, out);
}